// GliaGrow_60662118089240
// MI455X (gfx1250) — compile-verified
//
#include <hip/hip_runtime.h>

// ---------------------------------------------------------------------------
// Banded linear layer (band width 3) for MI455X / gfx1250.
//
// out[b, j] = x[b,j]*weight[j+2,j] + x[b,j+1]*weight[j+2,j+1]
//           + x[b,j+2]*weight[j+2,j+2] + bias[j+2]*count(j+2)
// for j in [0, F+2);  out is [4096, 8194] f32.
//
// Roofline: ~0.2 GFLOP vs ~256 MB of HBM traffic -> memory bound (~11 us at
// 23.3 TB/s). x (128 MB) fits in the 192 MB L2, so per-lane gathers are fine.
// Exact f32 math via V_WMMA_F32_16X16X4_F32: each 16x16 output tile is
// X[16x20] x W_band[20x16] as 5 chained K=4 WMMA steps; bias*count is folded
// into the accumulator init (it depends only on the output column, which is
// exactly what each lane owns in the C/D layout).
// ---------------------------------------------------------------------------

#define B_SIZE 4096
#define F_IN   8192
#define O_W    (F_IN + 2)          // 8194 output columns
#define NUM_JT ((O_W + 15) / 16)   // 513 column tiles
#define JT_PER_BLOCK 8             // 8 waves per 256-thread block (wave32)
#define BT_PER_WAVE  8             // batch tiles per wave; gridDim.y = 256/8 = 32

typedef float v2f __attribute__((ext_vector_type(2)));
typedef float v8f __attribute__((ext_vector_type(8)));

__global__ __launch_bounds__(256)
void band_linear_wmma_f32(const float* __restrict__ x,
                          const float* __restrict__ weight,
                          const float* __restrict__ bias,
                          float* __restrict__ out)
{
    const int lane = threadIdx.x & 31;
    const int wave = threadIdx.x >> 5;
    const int h    = lane >> 4;    // half-wave: selects K group (A,B) / M group (C,D)
    const int n    = lane & 15;    // output column within tile (B/C/D); batch row (A)

    // Column tile this wave owns. Clamp (not branch-out) so EXEC stays all-1s
    // for every WMMA; clamped waves redo tile 512 with identical results.
    int jt = blockIdx.x * JT_PER_BLOCK + wave;
    if (jt >= NUM_JT) jt = NUM_JT - 1;
    const int j0 = jt * 16;
    const int jn = j0 + n;                     // global output column of this lane

    // --- Band coefficients for output column jn (weight row c = jn+2) -----
    float w0 = 0.f, w1 = 0.f, w2 = 0.f, bb = 0.f;
    if (jn < F_IN)     w0 = weight[(long)(jn + 2) * F_IN + jn];       // * x[:,jn]
    if (jn + 1 < F_IN) w1 = weight[(long)(jn + 2) * F_IN + jn + 1];   // * x[:,jn+1]
    if (jn + 2 < F_IN) w2 = weight[(long)(jn + 2) * F_IN + jn + 2];   // * x[:,jn+2]
    if (jn < O_W) {
        float cnt = (float)((jn < F_IN) + (jn + 1 < F_IN) + (jn + 2 < F_IN));
        bb = bias[jn + 2] * cnt;               // bias re-added once per touch
    }

    // --- B fragments: W_band[20x16], 5 steps of K=4 -----------------------
    // 32-bit B layout (mirrors documented A 16x4 layout):
    //   VGPR p, half h -> row K = k0 + p + 2h, col N = lane%16.
    // W_band[k, n] = w0 if k==n, w1 if k==n+1, w2 if k==n+2, else 0.
    v2f bfrag[5];
#pragma unroll
    for (int s = 0; s < 5; ++s) {
#pragma unroll
        for (int p = 0; p < 2; ++p) {
            const int k = 4 * s + p + 2 * h;
            float w = 0.f;
            if (k == n)     w = w0;
            if (k == n + 1) w = w1;
            if (k == n + 2) w = w2;
            bfrag[s][p] = w;
        }
    }
    // Pin the fragments in VGPRs: without this the compiler rematerializes
    // the whole v_cmp/v_cndmask select chain before every WMMA in the
    // (unrolled) batch loop. 10 pinned VGPRs, computed once per wave.
    asm volatile("" : "+v"(bfrag[0]), "+v"(bfrag[1]), "+v"(bfrag[2]),
                      "+v"(bfrag[3]), "+v"(bfrag[4]));

    // Wave-uniform interior test: all 20 input columns j0..j0+19 in-bounds?
    // True for tiles 0..510; only the last two tiles need clamped loads.
    const bool interior = (j0 + 20 <= F_IN);

    // --- Batch-tile loop ---------------------------------------------------
    const int m = lane & 15;                   // A-matrix row = batch row in tile
    for (int bt = 0; bt < BT_PER_WAVE; ++bt) {
        const int b0 = (blockIdx.y * BT_PER_WAVE + bt) * 16;
        const float* xrow = x + (long)(b0 + m) * F_IN;

        v8f acc;                               // C/D: lane owns col n, rows v+8h
#pragma unroll
        for (int i = 0; i < 8; ++i) acc[i] = bb;

        if (interior) {
            // Fast path: one aligned b64 load per K-step (g is even -> 8B ok).
#pragma unroll
            for (int s = 0; s < 5; ++s) {
                // A 16x4 f32 layout: lane L: M=L%16; VGPR p holds K=p+2*(L/16).
                const v2f a = *(const v2f*)(xrow + j0 + 4 * s + 2 * h);
                // 8 args: (neg_a, A, neg_b, B, c_mod, C, reuse_a, reuse_b)
                acc = __builtin_amdgcn_wmma_f32_16x16x4_f32(
                    false, a, false, bfrag[s], (short)0, acc, false, false);
            }
        } else {
            // Edge path: clamp each element; W_band is 0 for any column >= F,
            // so the clamped (finite) x values multiply zeros.
#pragma unroll
            for (int s = 0; s < 5; ++s) {
                const int g  = j0 + 4 * s + 2 * h;
                const int i0 = (g     < F_IN) ? g     : F_IN - 1;
                const int i1 = (g + 1 < F_IN) ? g + 1 : F_IN - 1;
                v2f a;
                a[0] = xrow[i0];
                a[1] = xrow[i1];
                acc = __builtin_amdgcn_wmma_f32_16x16x4_f32(
                    false, a, false, bfrag[s], (short)0, acc, false, false);
            }
        }

        // D store: VGPR v, half h -> out[b0 + v + 8h][jn]; lanes 0-15 of each
        // VGPR cover 16 contiguous columns (64B), likewise lanes 16-31.
        if (jn < O_W) {
            float* op = out + (long)(b0 + 8 * h) * O_W + jn;
#pragma unroll
            for (int v = 0; v < 8; ++v)
                op[(long)v * O_W] = acc[v];
        }
    }
}

extern "C" void kernel_launch(void* const* d_in, const int* in_sizes, int n_in,
                              void* d_out, int out_size, void* d_ws, size_t ws_size,
                              hipStream_t stream) {
    (void)in_sizes; (void)n_in; (void)out_size; (void)d_ws; (void)ws_size;
    const float* x  = (const float*)d_in[0];   // [4096, 8192]
    const float* w  = (const float*)d_in[1];   // [8198, 8192]
    const float* bi = (const float*)d_in[2];   // [8198]
    float* out = (float*)d_out;                // [4096, 8194]

    dim3 grid((NUM_JT + JT_PER_BLOCK - 1) / JT_PER_BLOCK,   // 65 column-tile groups
              (B_SIZE / 16) / BT_PER_WAVE);                 // 32 batch strips
    dim3 block(256);                                        // 8 waves (wave32)
    hipLaunchKernelGGL(band_linear_wmma_f32, grid, block, 0, stream, x, w, bi, out);
}